// SimpleVQAutoEncoder_28604482191489
// MI455X (gfx1250) — compile-verified
//
#include <hip/hip_runtime.h>
#include <hip/hip_bf16.h>
#include <math.h>

typedef float v2f __attribute__((ext_vector_type(2)));
typedef float v8f __attribute__((ext_vector_type(8)));

#define A1_OFF   0            // a1 / g buffer: 128*16*64*64 = 8388608 floats
#define H_OFF    8388608      // h buffer:      128*32*32*32 = 4194304 floats
#define OUT_N    2097152      // 128*1*128*128
#define IDX_N    131072       // 128*32*32
#define LOSS_OFF (OUT_N + IDX_N)

__device__ __forceinline__ float gelu_exact(float v) {
    return 0.5f * v * (1.0f + erff(v * 0.70710678118654752f));
}

// ---------------- conv1 (1->16) + maxpool2 + gelu : x[128,1,128,128] -> a1[128,16,64,64]
__global__ __launch_bounds__(256) void k_enc1(const float* __restrict__ x,
                                              const float* __restrict__ w1,
                                              const float* __restrict__ b1,
                                              float* __restrict__ a1) {
    int o = blockIdx.x * 256 + threadIdx.x;           // 8388608 outputs
    int wq = o & 63, hq = (o >> 6) & 63, co = (o >> 12) & 15, b = o >> 16;
    const float* xin = x + (size_t)b * 128 * 128;
    float wv[9];
#pragma unroll
    for (int i = 0; i < 9; ++i) wv[i] = w1[co * 9 + i];
    float bias = b1[co];
    float m = -3.0e38f;
#pragma unroll
    for (int dy = 0; dy < 2; ++dy) {
#pragma unroll
        for (int dx = 0; dx < 2; ++dx) {
            int y = 2 * hq + dy, xx = 2 * wq + dx;
            float acc = bias;
#pragma unroll
            for (int ky = 0; ky < 3; ++ky) {
#pragma unroll
                for (int kx = 0; kx < 3; ++kx) {
                    int yy = y + ky - 1, xc = xx + kx - 1;
                    float v = (yy >= 0 && yy < 128 && xc >= 0 && xc < 128)
                                  ? xin[yy * 128 + xc] : 0.0f;
                    acc = fmaf(wv[ky * 3 + kx], v, acc);
                }
            }
            m = fmaxf(m, acc);
        }
    }
    a1[o] = gelu_exact(m);
}

// ---------------- conv2 (16->32) + maxpool2 : a1[128,16,64,64] -> h[128,32,32,32]
__global__ __launch_bounds__(256) void k_enc2(const float* __restrict__ a1,
                                              const float* __restrict__ w2,
                                              const float* __restrict__ b2,
                                              float* __restrict__ h) {
    int o = blockIdx.x * 256 + threadIdx.x;           // 4194304 outputs
    int wq = o & 31, hq = (o >> 5) & 31, co = (o >> 10) & 31, b = o >> 15;
    float acc0 = b2[co], acc1 = acc0, acc2 = acc0, acc3 = acc0;
    int y0 = 2 * hq, x0 = 2 * wq;
    for (int ci = 0; ci < 16; ++ci) {
        const float* win = w2 + (size_t)(co * 16 + ci) * 9;
        const float* ain = a1 + (size_t)(b * 16 + ci) * 64 * 64;
        float wv[9];
#pragma unroll
        for (int i = 0; i < 9; ++i) wv[i] = win[i];
#pragma unroll
        for (int dy = 0; dy < 2; ++dy) {
#pragma unroll
            for (int dx = 0; dx < 2; ++dx) {
                float s = 0.0f;
#pragma unroll
                for (int ky = 0; ky < 3; ++ky) {
#pragma unroll
                    for (int kx = 0; kx < 3; ++kx) {
                        int yy = y0 + dy + ky - 1, xc = x0 + dx + kx - 1;
                        float v = (yy >= 0 && yy < 64 && xc >= 0 && xc < 64)
                                      ? ain[yy * 64 + xc] : 0.0f;
                        s = fmaf(wv[ky * 3 + kx], v, s);
                    }
                }
                if (dy == 0 && dx == 0) acc0 += s;
                else if (dy == 0)       acc1 += s;
                else if (dx == 0)       acc2 += s;
                else                    acc3 += s;
            }
        }
    }
    h[o] = fmaxf(fmaxf(acc0, acc1), fmaxf(acc2, acc3));
}

// ---------------- VQ: nearest code via WMMA f32 16x16x4 GEMM + argmin + commit loss
// z[m][c] = h[(b*32+c)*1024 + sp], m = b*1024 + sp.  M=131072, N=512 codes, K=32 dims.
__global__ __launch_bounds__(256) void k_vq(const float* __restrict__ h,
                                            const float* __restrict__ codebook,
                                            float* __restrict__ idx_out,
                                            float* __restrict__ loss_out) {
    __shared__ float lds_cb[512 * 32];   // 64 KB codebook
    __shared__ float lds_cn[512];        // code norms
    int tid = threadIdx.x;

    // stage codebook into LDS (16384 floats, 16 x float4 per thread)
    const float4* cb4 = (const float4*)codebook;
    float4* l4 = (float4*)lds_cb;
#pragma unroll
    for (int i = 0; i < 16; ++i) l4[tid + 256 * i] = cb4[tid + 256 * i];
    __syncthreads();
#pragma unroll
    for (int i = 0; i < 2; ++i) {
        int k = tid * 2 + i;
        float s = 0.0f;
#pragma unroll
        for (int c = 0; c < 32; ++c) { float v = lds_cb[k * 32 + c]; s = fmaf(v, v, s); }
        lds_cn[k] = s;
    }
    __syncthreads();

    int lane = tid & 31;
    int wave = tid >> 5;
    int halfSel = lane >> 4;                 // 0: lanes 0-15, 1: lanes 16-31
    int rowBase = blockIdx.x * 128 + wave * 16;

    // A fragments per ISA 16x4 f32 layout: lanes 0-15 hold K=4kk+{0,1}, lanes 16-31 K=4kk+{2,3}
    int m = rowBase + (lane & 15);
    int b = m >> 10, sp = m & 1023;
    const float* zrow = h + (size_t)b * 32 * 1024 + sp;
    v2f a[8];
#pragma unroll
    for (int kk = 0; kk < 8; ++kk) {
        int c0 = kk * 4 + halfSel * 2;
        a[kk].x = zrow[(size_t)(c0 + 0) * 1024];
        a[kk].y = zrow[(size_t)(c0 + 1) * 1024];
    }

    float minv[8];
    int   mini[8];
#pragma unroll
    for (int r = 0; r < 8; ++r) { minv[r] = 3.0e38f; mini[r] = 0; }

    int col = lane & 15;
    for (int t = 0; t < 32; ++t) {           // 32 N-tiles of 16 codes
        int n0 = t * 16;
        v8f acc = {};
#pragma unroll
        for (int kk = 0; kk < 8; ++kk) {
            int kb = kk * 4 + halfSel * 2;
            v2f bf;
            bf.x = lds_cb[(n0 + col) * 32 + kb + 0];
            bf.y = lds_cb[(n0 + col) * 32 + kb + 1];
            acc = __builtin_amdgcn_wmma_f32_16x16x4_f32(
                false, a[kk], false, bf, (short)0, acc, false, false);
        }
        float cn = lds_cn[n0 + col];
#pragma unroll
        for (int r = 0; r < 8; ++r) {
            float d = cn - 2.0f * acc[r];    // ||c||^2 - 2 z.c  (||z||^2 constant per row)
            if (d < minv[r]) { minv[r] = d; mini[r] = n0 + col; }
        }
    }

    // 16-lane argmin reduction within each half-wave (D layout: VGPR r, lanes0-15=row r, lanes16-31=row r+8)
#pragma unroll
    for (int r = 0; r < 8; ++r) {
#pragma unroll
        for (int off = 1; off < 16; off <<= 1) {
            float ov = __shfl_xor(minv[r], off, 32);
            int   oi = __shfl_xor(mini[r], off, 32);
            if (ov < minv[r] || (ov == minv[r] && oi < mini[r])) { minv[r] = ov; mini[r] = oi; }
        }
    }
    if ((lane & 15) == 0) {
#pragma unroll
        for (int r = 0; r < 8; ++r)
            idx_out[rowBase + halfSel * 8 + r] = (float)mini[r];
    }

    // commit loss: wave-cooperative, lane = dim c
    float lsum = 0.0f;
    for (int j = 0; j < 16; ++j) {
        int src = (j < 8) ? 0 : 16;
        int code = __shfl(mini[j & 7], src, 32);
        int row = rowBase + j;
        int b2 = row >> 10, sp2 = row & 1023;
        float zv = h[(size_t)(b2 * 32 + lane) * 1024 + sp2];
        float d = lds_cb[code * 32 + lane] - zv;
        lsum = fmaf(d, d, lsum);
    }
#pragma unroll
    for (int off = 16; off >= 1; off >>= 1) lsum += __shfl_xor(lsum, off, 32);
    if (lane == 0) atomicAdd(loss_out, lsum * (1.0f / 4194304.0f)); // /(131072*32)
}

// ---------------- upsample2(h) -> conv3 (32->16) -> gelu : g[128,16,64,64]
__global__ __launch_bounds__(256) void k_dec1(const float* __restrict__ h,
                                              const float* __restrict__ w3,
                                              const float* __restrict__ b3,
                                              float* __restrict__ g) {
    int o = blockIdx.x * 256 + threadIdx.x;           // 8388608 outputs
    int wq = o & 63, hq = (o >> 6) & 63, co = (o >> 12) & 15, b = o >> 16;
    float acc = b3[co];
    for (int ci = 0; ci < 32; ++ci) {
        const float* win = w3 + (size_t)(co * 32 + ci) * 9;
        const float* hin = h + (size_t)(b * 32 + ci) * 32 * 32;
        float wv[9];
#pragma unroll
        for (int i = 0; i < 9; ++i) wv[i] = win[i];
#pragma unroll
        for (int ky = 0; ky < 3; ++ky) {
#pragma unroll
            for (int kx = 0; kx < 3; ++kx) {
                int yy = hq + ky - 1, xc = wq + kx - 1;
                float v = (yy >= 0 && yy < 64 && xc >= 0 && xc < 64)
                              ? hin[(yy >> 1) * 32 + (xc >> 1)] : 0.0f;
                acc = fmaf(wv[ky * 3 + kx], v, acc);
            }
        }
    }
    g[o] = gelu_exact(acc);
}

// ---------------- upsample2(g) -> conv4 (16->1) -> clip : out[128,1,128,128]
__global__ __launch_bounds__(256) void k_dec2(const float* __restrict__ g,
                                              const float* __restrict__ w4,
                                              const float* __restrict__ b4,
                                              float* __restrict__ out) {
    int o = blockIdx.x * 256 + threadIdx.x;           // 2097152 outputs
    int wq = o & 127, hq = (o >> 7) & 127, b = o >> 14;
    float acc = b4[0];
    for (int ci = 0; ci < 16; ++ci) {
        const float* win = w4 + (size_t)ci * 9;
        const float* gin = g + (size_t)(b * 16 + ci) * 64 * 64;
        float wv[9];
#pragma unroll
        for (int i = 0; i < 9; ++i) wv[i] = win[i];
#pragma unroll
        for (int ky = 0; ky < 3; ++ky) {
#pragma unroll
            for (int kx = 0; kx < 3; ++kx) {
                int yy = hq + ky - 1, xc = wq + kx - 1;
                float v = (yy >= 0 && yy < 128 && xc >= 0 && xc < 128)
                              ? gin[(yy >> 1) * 64 + (xc >> 1)] : 0.0f;
                acc = fmaf(wv[ky * 3 + kx], v, acc);
            }
        }
    }
    out[o] = fminf(1.0f, fmaxf(-1.0f, acc));
}

extern "C" void kernel_launch(void* const* d_in, const int* in_sizes, int n_in,
                              void* d_out, int out_size, void* d_ws, size_t ws_size,
                              hipStream_t stream) {
    const float* x        = (const float*)d_in[0];
    const float* w1       = (const float*)d_in[1];
    const float* b1       = (const float*)d_in[2];
    const float* w2       = (const float*)d_in[3];
    const float* b2       = (const float*)d_in[4];
    const float* codebook = (const float*)d_in[5];
    const float* w3       = (const float*)d_in[6];
    const float* b3       = (const float*)d_in[7];
    const float* w4       = (const float*)d_in[8];
    const float* b4       = (const float*)d_in[9];

    float* ws  = (float*)d_ws;
    float* a1  = ws + A1_OFF;             // reused as g after conv2 consumes it
    float* h   = ws + H_OFF;
    float* out = (float*)d_out;
    float* idx = out + OUT_N;
    float* los = out + LOSS_OFF;

    hipMemsetAsync(los, 0, sizeof(float), stream);

    k_enc1<<<32768, 256, 0, stream>>>(x, w1, b1, a1);
    k_enc2<<<16384, 256, 0, stream>>>(a1, w2, b2, h);
    k_vq  <<<1024,  256, 0, stream>>>(h, codebook, idx, los);
    k_dec1<<<32768, 256, 0, stream>>>(h, w3, b3, a1);   // a1 now holds g
    k_dec2<<<8192,  256, 0, stream>>>(a1, w4, b4, out);
}